// Kernel_Correlation_14267881357849
// MI455X (gfx1250) — compile-verified
//
#include <hip/hip_runtime.h>
#include <hip/hip_bf16.h>

typedef __attribute__((ext_vector_type(2))) float v2f;
typedef __attribute__((ext_vector_type(8))) float v8f;

#define SIGMA 0.2f

// One wave handles a 16-row tile of the [4n x 256] dot-product GEMM:
// rows = 16 consecutive (i,p) point rows (4 output points), cols = 256 kernel
// points, via 16x V_WMMA_F32_16X16X4_F32 (K=3 padded to 4 with a zero row).
// |p-k|^2 = |p|^2 - 2 p.k + |k|^2 ; resp = exp(-|p-k|^2 / (2 sigma^2)).
__global__ __launch_bounds__(128) void corr_wmma_kernel(
    const float* __restrict__ normal,        // [n,3]
    const long long* __restrict__ neighbour, // [n,3]
    const float* __restrict__ theta,         // [256]
    const float* __restrict__ phi,           // [256]
    float* __restrict__ fea,                 // [n,64]
    int n, int numTiles)
{
    const float C = 1.0f / (2.0f * SIGMA * SIGMA);   // 12.5

    // B operand pre-packed in WMMA lane layout:
    //   bpk[t][lane][0..1]  = lane<16 ? {kx[j], ky[j]} : {kz[j], 0},  j = 16t + (lane&15)
    // (B is 4x16 f32: VGPR0 = rows K=0/K=2, VGPR1 = rows K=1/K=3 across lane halves.)
    __shared__ float bpk[16][32][2];
    __shared__ float ksqC[256];              // |k|^2 * 12.5

    for (int j = threadIdx.x; j < 256; j += blockDim.x) {
        float th = theta[j], ph = phi[j];
        float st = __sinf(th), ct = __cosf(th);
        float sp = __sinf(ph), cp = __cosf(ph);
        float x = st * sp, y = st * cp, z = ct;
        int t = j >> 4, c = j & 15;
        bpk[t][c][0]      = x;    bpk[t][c][1]      = y;
        bpk[t][c + 16][0] = z;    bpk[t][c + 16][1] = 0.0f;
        ksqC[j] = (x * x + y * y + z * z) * C;
    }
    __syncthreads();

    const int lane = threadIdx.x & 31;
    const int wave = threadIdx.x >> 5;
    const int tile = blockIdx.x * 4 + wave;
    if (tile >= numTiles) return;

    const int r   = lane & 15;               // row within tile / N within col-tile
    const bool hi = lane >= 16;

    // ---- Load A tile: row = 16*tile + r  ->  (i = row/4, p = row%4) ----
    int row = tile * 16 + r;
    int last = 4 * n - 1;
    row = row > last ? last : row;           // clamp tail (stores guarded below)
    int i = row >> 2;
    int p = row & 3;
    long long src = (p == 0) ? (long long)i : neighbour[(long long)i * 3 + (p - 1)];
    float x = normal[src * 3 + 0];
    float y = normal[src * 3 + 1];
    float z = normal[src * 3 + 2];
    float psq = x * x + y * y + z * z;       // lane L holds |p|^2 of row (L&15)

    v2f a;                                   // A 16x4: lanes0-15 K=0,1 ; lanes16-31 K=2,3(=0)
    a.x = hi ? z : x;
    a.y = hi ? 0.0f : y;

    // Broadcast per-row |p|^2*12.5 for the 8 D rows this lane owns
    // (D reg rr holds row M = rr + (hi ? 8 : 0)).
    const int mbase = hi ? 8 : 0;
    float psqC[8];
#pragma unroll
    for (int rr = 0; rr < 8; ++rr)
        psqC[rr] = __shfl(psq, mbase + rr, 32) * C;

    const int i0 = tile * 4 + (hi ? 2 : 0);  // output point for D regs 0-3
    const int i1 = i0 + 1;                   // output point for D regs 4-7

    // ---- 16 column tiles of 16 kernel points each ----
    for (int t = 0; t < 16; ++t) {
        v2f b = *(const v2f*)&bpk[t][lane][0];        // one ds_load_b64, branch-free
        float kq = ksqC[t * 16 + r];                  // |k|^2 * 12.5 for N = 16t + r

        v8f d = {};
        d = __builtin_amdgcn_wmma_f32_16x16x4_f32(
                false, a, false, b, (short)0, d, false, false);

        // resp = exp(25*dot - (|p|^2 + |k|^2)*12.5)
        float s[8];
#pragma unroll
        for (int rr = 0; rr < 8; ++rr) s[rr] = psqC[rr] + kq;

        float slo = 0.0f, shi = 0.0f;
#pragma unroll
        for (int rr = 0; rr < 8; ++rr) {
            float e = __expf(__builtin_fmaf(d[rr], 25.0f, -s[rr]));
            if (rr < 4) slo += e; else shi += e;
        }
        // Sum over l: the 4 lanes of each quad share one m (N = 4m + l).
        slo += __shfl_xor(slo, 1, 32);
        slo += __shfl_xor(slo, 2, 32);
        shi += __shfl_xor(shi, 1, 32);
        shi += __shfl_xor(shi, 2, 32);

        if ((lane & 3) == 0) {
            int m = t * 4 + (r >> 2);
            if (i0 < n) fea[(long long)i0 * 64 + m] = slo * (1.0f / 16.0f);
            if (i1 < n) fea[(long long)i1 * 64 + m] = shi * (1.0f / 16.0f);
        }
    }
}

extern "C" void kernel_launch(void* const* d_in, const int* in_sizes, int n_in,
                              void* d_out, int out_size, void* d_ws, size_t ws_size,
                              hipStream_t stream) {
    const float*     normal    = (const float*)d_in[0];
    const long long* neighbour = (const long long*)d_in[1];
    const float*     theta     = (const float*)d_in[2];
    const float*     phi       = (const float*)d_in[3];
    float* fea = (float*)d_out;

    int n = in_sizes[0] / 3;                 // 50000
    int numTiles = (4 * n + 15) / 16;        // 12500 (16 point-rows per wave)
    int blocks = (numTiles + 3) / 4;         // 4 waves (128 threads) per block

    corr_wmma_kernel<<<blocks, 128, 0, stream>>>(normal, neighbour, theta, phi,
                                                 fea, n, numTiles);
}